// CARAFE_36129264894170
// MI455X (gfx1250) — compile-verified
//
#include <hip/hip_runtime.h>

// CARAFE-style upsampler for MI455X (gfx1250, wave32).
// Bandwidth-bound (~300MB HBM traffic @ 23.3 TB/s >> 8.9 GFLOPs) -> keep fp32
// and run the GEMM-shaped convs on V_WMMA_F32_16X16X4_F32 (matrix pipe, exact
// reference precision, no conversion traffic). The 105MB upsampled tensor is
// materialized in d_ws and stays L2-resident (192MB L2).
// Round-3 change: conv3x3 contraction reordered to k = e*Ci + ci so the tap
// offset/bounds mask is computed once per tap (9 unrolled) and the hot inner
// loop is branchless (mask-multiplied unconditional loads) -> no per-WMMA
// exec save/restore or div-by-9 magic sequences.

typedef __attribute__((ext_vector_type(2))) float v2f;
typedef __attribute__((ext_vector_type(8))) float v8f;

__device__ __forceinline__ float bn_silu(float y, float g, float b, float m, float v) {
    float inv = g * rsqrtf(v + 1e-5f);
    y = y * inv + (b - m * inv);
    return y / (1.0f + __expf(-y));   // y * sigmoid(y)
}

// ---------------------------------------------------------------------------
// 1x1 conv (GEMM) + BN + SiLU via V_WMMA_F32_16X16X4_F32.
// One wave computes MT 16(co)x16(pixel) tiles; K = Ci in steps of 4.
// A (16x4, weights): lanes 0-15 -> M rows, VGPR pair = K {0,1} (lanes 0-15)
//                    or K {2,3} (lanes 16-31).
// B (4x16, activations): lanes = pixels, same K-half split. B is loaded once
// per k-step and reused by all MT tiles (minimizes traffic on the big input).
// C/D: VGPR i -> M=i (lanes 0-15) / M=i+8 (lanes 16-31), N=lane&15.
// ---------------------------------------------------------------------------
template<int MT>
__global__ void conv1x1_bn_silu_wmma(
    const float* __restrict__ X,   // [B, Ci, P]
    const float* __restrict__ W,   // [MT*16, Ci]
    const float* __restrict__ g, const float* __restrict__ b,
    const float* __restrict__ m, const float* __restrict__ v,
    float* __restrict__ Y,         // [B, CtotOut, P], written at channel coOff
    int Ci, int P, int CtotOut, int coOff)
{
    const int lane  = threadIdx.x & 31;
    const int wave  = threadIdx.x >> 5;
    const int pix0  = (blockIdx.x * (blockDim.x >> 5) + wave) * 16;
    const int batch = blockIdx.y;

    const float* Xb = X + (size_t)batch * Ci * P;
    const int nn    = lane & 15;
    const int kHalf = (lane >> 4) * 2;            // 0 or 2
    const float* wr = W + (size_t)nn * Ci;        // row nn of tile 0

    v8f acc[MT];
#pragma unroll
    for (int t = 0; t < MT; ++t) acc[t] = (v8f){};

    for (int k0 = 0; k0 < Ci; k0 += 4) {
        const int k = k0 + kHalf;
        v2f bb;
        bb.x = Xb[(size_t)k * P + pix0 + nn];
        bb.y = Xb[(size_t)(k + 1) * P + pix0 + nn];
#pragma unroll
        for (int t = 0; t < MT; ++t) {
            v2f a;
            a.x = wr[(size_t)(t * 16) * Ci + k];
            a.y = wr[(size_t)(t * 16) * Ci + k + 1];
            acc[t] = __builtin_amdgcn_wmma_f32_16x16x4_f32(
                false, a, false, bb, (short)0, acc[t], false, false);
        }
    }

    const int pix   = pix0 + nn;
    const int mBase = (lane >> 4) * 8;
#pragma unroll
    for (int t = 0; t < MT; ++t) {
#pragma unroll
        for (int i = 0; i < 8; ++i) {
            const int co = t * 16 + mBase + i;
            const float y = bn_silu(acc[t][i], g[co], b[co], m[co], v[co]);
            Y[((size_t)batch * CtotOut + coOff + co) * P + pix] = y;
        }
    }
}

// ---------------------------------------------------------------------------
// 3x3 conv (implicit GEMM) + BN + SiLU via WMMA f32.
// Contraction order k = e*Ci + ci (A and B agree, sum is order-invariant):
// outer loop = 9 taps (unrolled; dy/dx constant, neighbor offset + bounds
// mask computed once per tap), inner loop = ci in steps of 4, fully
// branchless (clamped addresses + mask multiply).
// Output channels padded to a 16-multiple via clamped weight rows + masks.
// ---------------------------------------------------------------------------
__global__ void conv3x3_bn_silu_wmma(
    const float* __restrict__ X,   // [B, Ci, H, W]
    const float* __restrict__ W,   // [CoReal, Ci*9] (OIHW flattened)
    const float* __restrict__ g, const float* __restrict__ b,
    const float* __restrict__ m, const float* __restrict__ v,
    float* __restrict__ Y,         // [B, CtotOut, H, W] at channel coOff
    int Ci, int CoReal, int H, int Wd, int CtotOut, int coOff)
{
    const int lane  = threadIdx.x & 31;
    const int wave  = threadIdx.x >> 5;
    const int P     = H * Wd;
    const int pix0  = (blockIdx.x * (blockDim.x >> 5) + wave) * 16;
    const int batch = blockIdx.y;
    const int co0   = blockIdx.z * 16;

    const float* Xb = X + (size_t)batch * Ci * P;
    const int nn    = lane & 15;
    const int kHalf = (lane >> 4) * 2;            // 0 or 2
    const int pix   = pix0 + nn;
    const int ph    = pix / Wd;
    const int pw    = pix - ph * Wd;

    const int mrow   = co0 + nn;
    const int mclamp = (mrow < CoReal) ? mrow : (CoReal - 1);
    const float mmsk = (mrow < CoReal) ? 1.0f : 0.0f;
    const float* wr  = W + (size_t)mclamp * (size_t)(Ci * 9);

    v8f acc = {};
#pragma unroll
    for (int e = 0; e < 9; ++e) {
        const int dy = e / 3 - 1;                 // constants after unroll
        const int dx = e - (e / 3) * 3 - 1;
        const int hh = ph + dy, ww = pw + dx;
        const bool inb  = (hh >= 0) & (hh < H) & (ww >= 0) & (ww < Wd);
        const float xmsk = inb ? 1.0f : 0.0f;
        const float* xpe = Xb + (inb ? ((size_t)hh * Wd + ww) : (size_t)0);
        const float* wre = wr + e;

        for (int ci0 = 0; ci0 < Ci; ci0 += 4) {
            const int ci = ci0 + kHalf;
            v2f a, bb;
            a.x  = wre[(size_t)ci * 9]       * mmsk;
            a.y  = wre[(size_t)(ci + 1) * 9] * mmsk;
            bb.x = xpe[(size_t)ci * P]       * xmsk;
            bb.y = xpe[(size_t)(ci + 1) * P] * xmsk;
            acc = __builtin_amdgcn_wmma_f32_16x16x4_f32(
                false, a, false, bb, (short)0, acc, false, false);
        }
    }

    const int mBase = (lane >> 4) * 8;
#pragma unroll
    for (int i = 0; i < 8; ++i) {
        const int co = co0 + mBase + i;
        if (co < CoReal) {
            const float y = bn_silu(acc[i], g[co], b[co], m[co], v[co]);
            Y[((size_t)batch * CtotOut + coOff + co) * P + pix] = y;
        }
    }
}

// ---------------------------------------------------------------------------
// Depthwise 5x5 (pad 2) + BN + SiLU. Purely memory-bound; VALU.
// ---------------------------------------------------------------------------
__global__ void dw5x5_bn_silu(
    const float* __restrict__ X,   // [B, CtotIn, H, W], read at channel ciOff
    const float* __restrict__ Wt,  // [C, 1, 5, 5]
    const float* __restrict__ g, const float* __restrict__ b,
    const float* __restrict__ m, const float* __restrict__ v,
    float* __restrict__ Y,         // [B, CtotOut, H, W], write at channel coOff
    int B, int C, int H, int Wd, int CtotIn, int ciOff, int CtotOut, int coOff)
{
    const size_t idx   = (size_t)blockIdx.x * blockDim.x + threadIdx.x;
    const size_t total = (size_t)B * C * H * Wd;
    if (idx >= total) return;

    const int w = idx % Wd;
    const int h = (idx / Wd) % H;
    const int c = (int)((idx / ((size_t)Wd * H)) % C);
    const int n = (int)(idx / ((size_t)Wd * H * C));

    const float* xp = X + ((size_t)n * CtotIn + ciOff + c) * H * Wd;
    const float* wp = Wt + c * 25;

    // Prefetch the row below the 5x5 window (helps the streaming pattern).
    if (h + 3 < H) __builtin_prefetch(&xp[(size_t)(h + 3) * Wd + w], 0, 1);

    float s = 0.0f;
#pragma unroll
    for (int dy = 0; dy < 5; ++dy) {
        const int hh = h + dy - 2;
        if (hh < 0 || hh >= H) continue;
#pragma unroll
        for (int dx = 0; dx < 5; ++dx) {
            const int ww = w + dx - 2;
            if (ww < 0 || ww >= Wd) continue;
            s += xp[(size_t)hh * Wd + ww] * wp[dy * 5 + dx];
        }
    }
    const float y = bn_silu(s, g[c], b[c], m[c], v[c]);
    Y[((size_t)n * CtotOut + coOff + c) * H * Wd + (size_t)h * Wd + w] = y;
}

// ---------------------------------------------------------------------------
// Softmax over the 9 kernel taps. kt: [B, 36, P] with channel = kk*4 + r.
// Output wgt: [B, P, 9, 4].
// ---------------------------------------------------------------------------
__global__ void softmax9(const float* __restrict__ KT, float* __restrict__ Wg,
                         int B, int P)
{
    const int idx = blockIdx.x * blockDim.x + threadIdx.x;
    if (idx >= B * P * 4) return;
    const int r = idx & 3;
    const int p = (idx >> 2) % P;
    const int n = (idx >> 2) / P;

    const float* kp = KT + (size_t)n * 36 * P + p;
    float vals[9];
    float mx = -3.0e38f;
#pragma unroll
    for (int kk = 0; kk < 9; ++kk) {
        vals[kk] = kp[(size_t)(kk * 4 + r) * P];
        mx = fmaxf(mx, vals[kk]);
    }
    float sum = 0.0f;
#pragma unroll
    for (int kk = 0; kk < 9; ++kk) {
        vals[kk] = __expf(vals[kk] - mx);
        sum += vals[kk];
    }
    const float inv = 1.0f / sum;
    float* wp = Wg + ((size_t)n * P + p) * 36 + r;
#pragma unroll
    for (int kk = 0; kk < 9; ++kk) wp[kk * 4] = vals[kk] * inv;
}

// ---------------------------------------------------------------------------
// CARAFE reassembly + 2x pixel shuffle, LDS-staged weights.
// o[n,c,2h+r1,2w+r2] = sum_e x[n,c,h+dy,w+dx] * wgt[n,h,w,e,r]
// One block owns 64 consecutive pixels of one image: the 64x36 weight block is
// staged once in LDS (9KB), hoisted to registers, then reused for all C
// channels. Threads: 64 pixels x 4 channel-slices.
// ---------------------------------------------------------------------------
__global__ void carafe_kernel(const float* __restrict__ X,
                              const float* __restrict__ Wg,
                              float* __restrict__ O,
                              int B, int C, int H, int Wd)
{
    __shared__ float sw[64 * 36];

    const int P  = H * Wd;
    const int n  = blockIdx.y;
    const int p0 = blockIdx.x * 64;

    // Cooperative stage of the weight block: 2304 floats, 256 threads.
    const float* wsrc = Wg + ((size_t)n * P + p0) * 36;
    for (int i = threadIdx.x; i < 64 * 36; i += 256) sw[i] = wsrc[i];
    __syncthreads();

    const int pl  = threadIdx.x & 63;     // pixel within tile
    const int cl  = threadIdx.x >> 6;     // channel slice 0..3
    const int pix = p0 + pl;
    const int h   = pix / Wd;
    const int w   = pix - h * Wd;

    // Hoist this pixel's 36 weights to registers (read LDS once).
    float wv[36];
#pragma unroll
    for (int i = 0; i < 36; ++i) wv[i] = sw[pl * 36 + i];

    const int H2 = H * 2, W2 = Wd * 2;
    const size_t obase = (size_t)(2 * h) * W2 + 2 * w;

    for (int c = cl; c < C; c += 4) {
        const float* xp = X + ((size_t)n * C + c) * P;
        float xv[9];
#pragma unroll
        for (int e = 0; e < 9; ++e) {
            const int dy = e / 3 - 1, dx = e % 3 - 1;
            const int hh = h + dy, ww = w + dx;
            const bool inb = (hh >= 0) & (hh < H) & (ww >= 0) & (ww < Wd);
            xv[e] = inb ? xp[(size_t)hh * Wd + ww] : 0.0f;
        }
        float* op = O + ((size_t)n * C + c) * H2 * W2;
#pragma unroll
        for (int r = 0; r < 4; ++r) {
            float s = 0.0f;
#pragma unroll
            for (int e = 0; e < 9; ++e) s += xv[e] * wv[e * 4 + r];
            op[obase + (size_t)(r >> 1) * W2 + (r & 1)] = s;
        }
    }
}

// ---------------------------------------------------------------------------
// Launch: inputs in setup_inputs() order (x, then 5 params per conv, 6 convs).
// ---------------------------------------------------------------------------
extern "C" void kernel_launch(void* const* d_in, const int* in_sizes, int n_in,
                              void* d_out, int out_size, void* d_ws, size_t ws_size,
                              hipStream_t stream)
{
    (void)in_sizes; (void)n_in; (void)out_size; (void)ws_size;
    const float* x = (const float*)d_in[0];
    #define CONVP(i) (const float*)d_in[i], (const float*)d_in[i+1], (const float*)d_in[i+2], \
                     (const float*)d_in[i+3], (const float*)d_in[i+4]

    float* ws   = (float*)d_ws;
    float* cat1 = ws;                                  // [4, 64, 80, 80]
    float* kt   = cat1 + (size_t)4 * 64 * 6400;        // [4, 36, 80, 80]
    float* wgt  = kt   + (size_t)4 * 36 * 6400;        // [4, 6400, 9, 4]
    float* o    = wgt  + (size_t)4 * 6400 * 36;        // [4, 256, 160, 160]
    float* out  = (float*)d_out;                       // [4, 256, 160, 160]

    const dim3 wblk(128);  // 4 waves; each wave = MT WMMA tiles of 16 pixels

    // 1) down_cv1: 1x1 conv 256->32 (MT=2), write cat1[:, 0:32]
    conv1x1_bn_silu_wmma<2><<<dim3(100, 4), wblk, 0, stream>>>(
        x, CONVP(1), cat1, 256, 6400, 64, 0);

    // 2) down_cv2: depthwise 5x5 on 32ch, cat1[:,0:32] -> cat1[:,32:64]
    {
        const int nthr = 4 * 32 * 6400;
        dw5x5_bn_silu<<<(nthr + 255) / 256, 256, 0, stream>>>(
            cat1, CONVP(6), cat1, 4, 32, 80, 80, 64, 0, 64, 32);
    }

    // 3) enc_cv1: 3x3 conv 64->18 (pad 1), write kt[:, 0:18]
    conv3x3_bn_silu_wmma<<<dim3(100, 4, 2), wblk, 0, stream>>>(
        cat1, CONVP(11), kt, 64, 18, 80, 80, 36, 0);

    // 4) enc_cv2: depthwise 5x5 on 18ch, kt[:,0:18] -> kt[:,18:36]
    {
        const int nthr = 4 * 18 * 6400;
        dw5x5_bn_silu<<<(nthr + 255) / 256, 256, 0, stream>>>(
            kt, CONVP(16), kt, 4, 18, 80, 80, 36, 0, 36, 18);
    }

    // 5) softmax over 9 taps -> wgt [4, 6400, 9, 4]
    {
        const int nthr = 4 * 6400 * 4;
        softmax9<<<(nthr + 255) / 256, 256, 0, stream>>>(kt, wgt, 4, 6400);
    }

    // 6) CARAFE reassembly + pixel shuffle -> o [4, 256, 160, 160] (L2-resident)
    carafe_kernel<<<dim3(100, 4), 256, 0, stream>>>(x, wgt, o, 4, 256, 80, 80);

    // 7) out_cv1: 1x1 conv 256->128 over 160x160 (MT=8, input read once),
    //    write d_out[:, 0:128]
    conv1x1_bn_silu_wmma<8><<<dim3(400, 4), wblk, 0, stream>>>(
        o, CONVP(21), out, 256, 25600, 256, 0);

    // 8) out_cv2: depthwise 5x5 on 128ch, d_out[:,0:128] -> d_out[:,128:256]
    {
        const int nthr = 4 * 128 * 25600;
        dw5x5_bn_silu<<<(nthr + 255) / 256, 256, 0, stream>>>(
            out, CONVP(26), out, 4, 128, 160, 160, 256, 0, 256, 128);
    }
    #undef CONVP
}